// CustomLinear_66005057405513
// MI455X (gfx1250) — compile-verified
//
#include <hip/hip_runtime.h>

// CDNA5 / gfx1250 batched 4096-point complex FFT with pre-twiddle, radix-16
// Cooley-Tukey (4096 = 16*16*16). Each 16-point butterfly bank is a complex
// 16x16 x 16x16 matmul executed on the f32 WMMA pipe as 4 K=4 chunks.
//
// One workgroup (256 threads = 8 wave32) per batch; signal lives in LDS
// (2 x 16 KB, split re/im). Stages run in place; inter-stage twiddles applied
// on the WMMA accumulator before the LDS store.

typedef __attribute__((ext_vector_type(2))) float v2f;
typedef __attribute__((ext_vector_type(8))) float v8f;

#define FFT_N     4096
#define OUT_HALF  3072
#define TWO_PI    6.283185307179586f

__device__ __forceinline__ v8f wmma4(v2f a, v2f b, v8f c) {
  // V_WMMA_F32_16X16X4_F32 : D[16x16] = A[16x4] * B[4x16] + C
  // 8-arg pattern: (neg_a, A, neg_b, B, c_mod, C, reuse_a, reuse_b)
  return __builtin_amdgcn_wmma_f32_16x16x4_f32(
      false, a, false, b, (short)0, c, false, false);
}

// STAGE 1: stride 256 butterflies, twiddle w4096^(a*j), a = 16*tile + m
// STAGE 2: stride 16 butterflies (block=tile), twiddle w256^(m*j)
// STAGE 3: stride 1 butterflies, no twiddle
template <int STAGE>
__device__ __forceinline__ void fft_stage(float* lr, float* li, int tile,
                                          int m, int h,
                                          const v2f* Fr, const v2f* Fi,
                                          const v2f* Fni) {
  int base, kstride;
  if (STAGE == 1)      { base = 16 * tile + m;        kstride = 256; }
  else if (STAGE == 2) { base = m + 256 * tile;       kstride = 16;  }
  else                 { base = 256 * tile + 16 * m;  kstride = 1;   }

  // Load B operand chunks: X[k][n], n = m (this lane's column), k = 4c+2h,+1
  v2f Xr[4], Xi[4];
#pragma unroll
  for (int c = 0; c < 4; ++c) {
    int i0 = base + (4 * c + 2 * h) * kstride;
    int i1 = i0 + kstride;
    Xr[c].x = lr[i0]; Xr[c].y = lr[i1];
    Xi[c].x = li[i0]; Xi[c].y = li[i1];
  }

  // Complex matmul: Yr = Fr*Xr - Fi*Xi ; Yi = Fr*Xi + Fi*Xr
  v8f Yr = {0.f, 0.f, 0.f, 0.f, 0.f, 0.f, 0.f, 0.f};
  v8f Yi = {0.f, 0.f, 0.f, 0.f, 0.f, 0.f, 0.f, 0.f};
#pragma unroll
  for (int c = 0; c < 4; ++c) {
    Yr = wmma4(Fr[c],  Xr[c], Yr);
    Yr = wmma4(Fni[c], Xi[c], Yr);
    Yi = wmma4(Fr[c],  Xi[c], Yi);
    Yi = wmma4(Fi[c],  Xr[c], Yi);
  }

  // Twiddle + store. D reg v holds row j = v + 8*h, col n = m.
#pragma unroll
  for (int v = 0; v < 8; ++v) {
    int j = v + 8 * h;
    float yr = Yr[v], yi = Yi[v];
    int widx = base + j * kstride;
    if (STAGE == 1) {
      int a = 16 * tile + m;
      float ang = -(float)(a * j) * (TWO_PI / 4096.0f);
      float s, c;
      __sincosf(ang, &s, &c);
      lr[widx] = yr * c - yi * s;
      li[widx] = yr * s + yi * c;
    } else if (STAGE == 2) {
      float ang = -(float)(m * j) * (TWO_PI / 256.0f);
      float s, c;
      __sincosf(ang, &s, &c);
      lr[widx] = yr * c - yi * s;
      li[widx] = yr * s + yi * c;
    } else {
      lr[widx] = yr;
      li[widx] = yi;
    }
  }
}

__global__ __launch_bounds__(256) void fft4096_wmma_kernel(
    const float2* __restrict__ x, const float2* __restrict__ w,
    float2* __restrict__ out) {
  __shared__ float lr[FFT_N];
  __shared__ float li[FFT_N];

  const int tid   = threadIdx.x;
  const int batch = blockIdx.x;
  const int lane  = tid & 31;       // wave32
  const int wave  = tid >> 5;       // 8 waves
  const int m     = lane & 15;      // A row / B,D column for this lane
  const int h     = lane >> 4;      // half-wave selector

  // Persistent DFT16 operand chunks (A-matrix layout, K in 4-chunks):
  // F[j][k] = exp(-2*pi*i*j*k/16); lane holds rows m, cols 4c+2h, 4c+2h+1.
  v2f Fr[4], Fi[4], Fni[4];
#pragma unroll
  for (int c = 0; c < 4; ++c) {
    int k0 = 4 * c + 2 * h;
    float a0 = -(float)((m * k0) & 15) * (TWO_PI / 16.0f);
    float a1 = -(float)((m * (k0 + 1)) & 15) * (TWO_PI / 16.0f);
    float s0, c0, s1, c1;
    __sincosf(a0, &s0, &c0);
    __sincosf(a1, &s1, &c1);
    Fr[c].x  = c0;  Fr[c].y  = c1;
    Fi[c].x  = s0;  Fi[c].y  = s1;
    Fni[c].x = -s0; Fni[c].y = -s1;
  }

  // Load + pre-twiddle: lds = x * w (complex), split re/im.
  const float2* xb = x + (size_t)batch * FFT_N;
  for (int i = tid; i < FFT_N; i += 256) {
    float2 xv = xb[i];
    float2 wv = w[i];
    lr[i] = xv.x * wv.x - xv.y * wv.y;
    li[i] = xv.x * wv.y + xv.y * wv.x;
  }
  __syncthreads();

  // 3 radix-16 stages; 16 tiles per stage, 2 tiles per wave, in place.
#pragma unroll
  for (int t = 0; t < 2; ++t)
    fft_stage<1>(lr, li, 2 * wave + t, m, h, Fr, Fi, Fni);
  __syncthreads();
#pragma unroll
  for (int t = 0; t < 2; ++t)
    fft_stage<2>(lr, li, 2 * wave + t, m, h, Fr, Fi, Fni);
  __syncthreads();
#pragma unroll
  for (int t = 0; t < 2; ++t)
    fft_stage<3>(lr, li, 2 * wave + t, m, h, Fr, Fi, Fni);
  __syncthreads();

  // Digit-reversed gather of first OUT_HALF bins:
  // X[f0 + 16*f1 + 256*f2] lives at lds[f2 + 16*f1 + 256*f0].
  float2* ob = out + (size_t)batch * OUT_HALF;
  for (int f = tid; f < OUT_HALF; f += 256) {
    int f0 = f & 15, f1 = (f >> 4) & 15, f2 = f >> 8;
    int idx = f2 + 16 * f1 + 256 * f0;
    float2 r;
    r.x = lr[idx];
    r.y = li[idx];
    ob[f] = r;
  }
}

extern "C" void kernel_launch(void* const* d_in, const int* in_sizes, int n_in,
                              void* d_out, int out_size, void* d_ws,
                              size_t ws_size, hipStream_t stream) {
  (void)in_sizes; (void)n_in; (void)d_ws; (void)ws_size; (void)out_size;
  const float2* x = (const float2*)d_in[0];  // [4096, 4096, 2] f32
  const float2* w = (const float2*)d_in[1];  // [2*4096] f32
  float2* out = (float2*)d_out;              // [4096, 3072] complex
  hipLaunchKernelGGL(fft4096_wmma_kernel, dim3(4096), dim3(256), 0, stream,
                     x, w, out);
}